// HexaToParallelogram_33578054320625
// MI455X (gfx1250) — compile-verified
//
#include <hip/hip_runtime.h>
#include <cstdint>
#include <cstddef>

#define NPIX         1039      // NUM_PIXELS from the reference
#define ROWS_PER_BLK 4         // 4*1039 floats = 16624 B chunk, 16B-aligned per block
#define MAX_GRID     1536      // >= 31*39 = 1209
#define BLOCK        256       // 8 waves (wave32) per block

typedef int v4i __attribute__((vector_size(16)));

typedef __attribute__((address_space(1))) int*  gas1_ip;
typedef __attribute__((address_space(3))) int*  as3_ip;
typedef __attribute__((address_space(1))) v4i*  gas1_v4ip;
typedef __attribute__((address_space(3))) v4i*  as3_v4ip;

// ---- CDNA5 async global->LDS copies ------------------------------------
__device__ __forceinline__ void g2l_async_b32(const void* gsrc, void* ldst) {
#if defined(__gfx1250__) && __has_builtin(__builtin_amdgcn_global_load_async_to_lds_b32)
    __builtin_amdgcn_global_load_async_to_lds_b32(
        (gas1_ip)gsrc, (as3_ip)ldst, /*offset=*/0, /*cpol=*/0);
#else
    *(unsigned int*)ldst = *(const unsigned int*)gsrc;   // fallback
#endif
}

__device__ __forceinline__ void g2l_async_b128(const void* gsrc, void* ldst) {
#if defined(__gfx1250__) && __has_builtin(__builtin_amdgcn_global_load_async_to_lds_b128)
    __builtin_amdgcn_global_load_async_to_lds_b128(
        (gas1_v4ip)gsrc, (as3_v4ip)ldst, /*offset=*/0, /*cpol=*/0);
#else
    g2l_async_b32((const char*)gsrc +  0, (char*)ldst +  0);
    g2l_async_b32((const char*)gsrc +  4, (char*)ldst +  4);
    g2l_async_b32((const char*)gsrc +  8, (char*)ldst +  8);
    g2l_async_b32((const char*)gsrc + 12, (char*)ldst + 12);
#endif
}

__device__ __forceinline__ void wait_async_zero() {
#if defined(__gfx1250__) && __has_builtin(__builtin_amdgcn_s_wait_asynccnt)
    __builtin_amdgcn_s_wait_asynccnt(0);
#elif defined(__gfx1250__)
    asm volatile("s_wait_asynccnt 0" ::: "memory");
#endif
}

__global__ __launch_bounds__(BLOCK)
void hexa_to_parallelogram_kernel(const float* __restrict__ hexa,
                                  const int* __restrict__ gidx,
                                  float* __restrict__ out,
                                  int n_grid, int rows_total) {
    __shared__ __align__(16) float s_row[ROWS_PER_BLK * NPIX];
    __shared__ int s_idx[MAX_GRID];

    const int tid       = threadIdx.x;
    const int row0      = blockIdx.x * ROWS_PER_BLK;
    const int rows_this = min(ROWS_PER_BLK, rows_total - row0);

    // ---- Stage gather map into LDS (async DMA path, B32) ----
    for (int g = tid; g < n_grid; g += BLOCK)
        g2l_async_b32(&gidx[g], &s_idx[g]);

    // ---- Stage ROWS_PER_BLK contiguous hexa rows as one chunk (B128) ----
    {
        const float* src   = hexa + (size_t)row0 * NPIX;   // 16B-aligned chunk
        const int    elems = rows_this * NPIX;
        const int    nvec  = elems >> 2;                   // 4*1039 -> 1039 exactly
        for (int v = tid; v < nvec; v += BLOCK)
            g2l_async_b128(&src[v * 4], &s_row[v * 4]);
        for (int c = (nvec << 2) + tid; c < elems; c += BLOCK)   // ragged tail
            g2l_async_b32(&src[c], &s_row[c]);
    }

    wait_async_zero();     // this wave's async LDS writes are visible
    __syncthreads();       // ... and so are every other wave's

    // ---- Gather + pad, streaming NT stores (write-once output) ----
    if (rows_this == ROWS_PER_BLK) {
        for (int g = tid; g < n_grid; g += BLOCK) {
            const int  i     = s_idx[g];
            const bool valid = (i >= 0);
            float* o = out + (size_t)row0 * n_grid + g;
#pragma unroll
            for (int r = 0; r < ROWS_PER_BLK; ++r) {
                float v = valid ? s_row[r * NPIX + i] : 0.0f;
                __builtin_nontemporal_store(v, o + (size_t)r * n_grid);
            }
        }
    } else {
        for (int g = tid; g < n_grid; g += BLOCK) {
            const int  i     = s_idx[g];
            const bool valid = (i >= 0);
            for (int r = 0; r < rows_this; ++r) {
                float v = valid ? s_row[r * NPIX + i] : 0.0f;
                __builtin_nontemporal_store(
                    v, &out[(size_t)(row0 + r) * n_grid + g]);
            }
        }
    }
}

extern "C" void kernel_launch(void* const* d_in, const int* in_sizes, int n_in,
                              void* d_out, int out_size, void* d_ws, size_t ws_size,
                              hipStream_t stream) {
    const float* hexa = (const float*)d_in[0];
    const int*   gidx = (const int*)d_in[1];
    float*       out  = (float*)d_out;

    const int n_grid     = in_sizes[1];          // 31*39 = 1209 map cells
    const int rows_total = in_sizes[0] / NPIX;   // 32768 batch rows

    const int blocks = (rows_total + ROWS_PER_BLK - 1) / ROWS_PER_BLK;
    hipLaunchKernelGGL(hexa_to_parallelogram_kernel,
                       dim3(blocks), dim3(BLOCK), 0, stream,
                       hexa, gidx, out, n_grid, rows_total);
}